// Edge2NodeProp2_15152644620441
// MI455X (gfx1250) — compile-verified
//
#include <hip/hip_runtime.h>
#include <cstdint>
#include <cstddef>

#define N_EDGE_C   1000000
#define N_NODE_C   100000
#define EDGE_DIM_C 64
#define N_RADIAL_C 16
#define UP_DIM_C   256
#define MLP_ROWS   64          // nodes per block in the MLP kernel
#define LDH        260         // LDS row stride (floats): even (8B align) and %64==4 -> conflict-free A frags

typedef float v2f __attribute__((ext_vector_type(2)));
typedef float v8f __attribute__((ext_vector_type(8)));

// ---------------------------------------------------------------------------
// Stage 0: zero the node accumulator
// ---------------------------------------------------------------------------
__global__ void zero_f32(float* __restrict__ p, int n) {
    int i = blockIdx.x * blockDim.x + threadIdx.x;
    if (i < n) p[i] = 0.0f;
}

// ---------------------------------------------------------------------------
// Stage 1: e = (rbf @ W_rbf) * x, scatter-add into h0[node][64]
// One thread per (edge, col). Block = 4 edges x 64 cols.
// ---------------------------------------------------------------------------
__global__ void gate_scatter(const float* __restrict__ x,
                             const float* __restrict__ rbf,
                             const long long* __restrict__ idx,
                             const float* __restrict__ W_rbf,
                             float* __restrict__ h0) {
    __shared__ float sW[N_RADIAL_C * EDGE_DIM_C];  // 16x64 = 4 KB
    __shared__ float srbf[4 * N_RADIAL_C];         // 4 edges' rbf rows

    const int tid = threadIdx.x;

    // cooperative load of W_rbf (1024 floats / 256 threads)
#pragma unroll
    for (int i = 0; i < 4; ++i) sW[tid + i * 256] = W_rbf[tid + i * 256];

    // 4 rbf rows for this block's edges
    if (tid < 64) {
        int e = tid >> 4;          // 0..3
        int k = tid & 15;
        srbf[e * 16 + k] = rbf[((long long)blockIdx.x * 4 + e) * N_RADIAL_C + k];
    }
    __syncthreads();

    const long long t    = (long long)blockIdx.x * 256 + tid;  // global (edge,col) id
    const int       edge = (int)(t >> 6);
    const int       col  = tid & 63;
    const int       eloc = tid >> 6;                           // 0..3

    float a = 0.0f;
#pragma unroll
    for (int k = 0; k < N_RADIAL_C; ++k)
        a += srbf[eloc * 16 + k] * sW[k * EDGE_DIM_C + col];

    const float v    = a * x[t];
    const int   node = (int)idx[edge];
    unsafeAtomicAdd(&h0[(size_t)node * EDGE_DIM_C + col], v);
}

// ---------------------------------------------------------------------------
// Fused node MLP: one block handles 64 nodes, activations live in LDS.
// GEMMs use V_WMMA_F32_16X16X4_F32 (fp32 in/out, matches reference precision).
// ---------------------------------------------------------------------------
__device__ __forceinline__ void gemm_layer(const float* __restrict__ Hin,   // LDS, [64][LDH]
                                           float* __restrict__ Hout,        // LDS, [64][LDH]
                                           const float* __restrict__ W,     // global, [K][256]
                                           const float* __restrict__ bias,  // global, [256] or null
                                           int K, bool act, int lane, int wave) {
    const int n_l   = lane & 15;   // col within 16-wide tile / row within M-tile
    const int khalf = lane >> 4;   // 0: K pair {kb,kb+1}; 1: {kb+2,kb+3}

    // 16 N-tiles of width 16; each of the 8 waves owns 2
    for (int t = wave; t < 16; t += 8) {
        const int n0 = t * 16;
        v8f acc[4] = {v8f{0}, v8f{0}, v8f{0}, v8f{0}};

#pragma unroll 4
        for (int kb = 0; kb < K; kb += 4) {
            const int kr = kb + 2 * khalf;
            // B fragment: rows kr, kr+1 of W at cols [n0, n0+16)
            v2f bf;
            bf.x = W[(size_t)kr * UP_DIM_C + n0 + n_l];
            bf.y = W[(size_t)(kr + 1) * UP_DIM_C + n0 + n_l];
#pragma unroll
            for (int mt = 0; mt < 4; ++mt) {
                const float* ap = &Hin[(mt * 16 + n_l) * LDH + kr];
                v2f af;
                af.x = ap[0];
                af.y = ap[1];
                acc[mt] = __builtin_amdgcn_wmma_f32_16x16x4_f32(
                    false, af, false, bf, (short)0, acc[mt], false, false);
            }
        }

        const float bv = bias ? bias[n0 + n_l] : 0.0f;
#pragma unroll
        for (int mt = 0; mt < 4; ++mt) {
#pragma unroll
            for (int r = 0; r < 8; ++r) {
                float v = acc[mt][r] + bv;
                if (act) v = v / (1.0f + __expf(-v));  // SiLU
                const int ml = mt * 16 + r + 8 * khalf;
                Hout[ml * LDH + n0 + n_l] = v;
            }
        }
    }
}

__global__ void node_mlp(const float* __restrict__ h0,
                         const float* __restrict__ Wup,
                         const float* __restrict__ W1, const float* __restrict__ b1,
                         const float* __restrict__ W2, const float* __restrict__ b2,
                         const float* __restrict__ W3, const float* __restrict__ b3,
                         const float* __restrict__ Wout,
                         float* __restrict__ out) {
    extern __shared__ float smem[];
    float* bufA = smem;
    float* bufB = smem + MLP_ROWS * LDH;

    const int tid  = threadIdx.x;
    const int lane = tid & 31;
    const int wave = tid >> 5;
    const int base = blockIdx.x * MLP_ROWS;

    // Load H0 tile (64 x 64) into bufA; rows past N_NODE are zero.
    {
        const int  row   = tid >> 2;
        const int  cq    = (tid & 3) * 16;
        const bool valid = (base + row) < N_NODE_C;
#pragma unroll
        for (int i = 0; i < 4; ++i) {
            float4 v = valid
                           ? *(const float4*)&h0[(size_t)(base + row) * EDGE_DIM_C + cq + i * 4]
                           : make_float4(0.f, 0.f, 0.f, 0.f);
            *(float4*)&bufA[row * LDH + cq + i * 4] = v;
        }
    }
    __syncthreads();

    gemm_layer(bufA, bufB, Wup, nullptr, EDGE_DIM_C, false, lane, wave);  // h @ W_up
    __syncthreads();
    gemm_layer(bufB, bufA, W1, b1, UP_DIM_C, true, lane, wave);           // silu(. @ W1 + b1)
    __syncthreads();
    gemm_layer(bufA, bufB, W2, b2, UP_DIM_C, true, lane, wave);           // silu(. @ W2 + b2)
    __syncthreads();
    gemm_layer(bufB, bufA, W3, b3, UP_DIM_C, true, lane, wave);           // silu(. @ W3 + b3)
    __syncthreads();

    // Final projection: out = H @ W_out  (256 -> 1)
    if (tid < MLP_ROWS) {
        const int node = base + tid;
        if (node < N_NODE_C) {
            float s = 0.0f;
#pragma unroll 8
            for (int c = 0; c < UP_DIM_C; ++c) s += bufA[tid * LDH + c] * Wout[c];
            out[node] = s;
        }
    }
}

// ---------------------------------------------------------------------------
extern "C" void kernel_launch(void* const* d_in, const int* in_sizes, int n_in,
                              void* d_out, int out_size, void* d_ws, size_t ws_size,
                              hipStream_t stream) {
    const float*     x     = (const float*)d_in[0];
    const float*     rbf   = (const float*)d_in[1];
    const long long* idx   = (const long long*)d_in[2];
    // d_in[3] = num_nodes (compile-time constant here)
    const float* W_rbf = (const float*)d_in[4];
    const float* W_up  = (const float*)d_in[5];
    const float* W1    = (const float*)d_in[6];
    const float* b1    = (const float*)d_in[7];
    const float* W2    = (const float*)d_in[8];
    const float* b2    = (const float*)d_in[9];
    const float* W3    = (const float*)d_in[10];
    const float* b3    = (const float*)d_in[11];
    const float* W_out = (const float*)d_in[12];
    float*       out   = (float*)d_out;
    float*       h0    = (float*)d_ws;  // 100000 x 64 fp32 = 25.6 MB scratch

    // Stage 0: zero accumulator
    const int nh = N_NODE_C * EDGE_DIM_C;
    zero_f32<<<(nh + 255) / 256, 256, 0, stream>>>(h0, nh);

    // Stage 1: gate + scatter-add (1M edges x 64 cols, 4 edges per block)
    gate_scatter<<<(N_EDGE_C * EDGE_DIM_C) / 256, 256, 0, stream>>>(x, rbf, idx, W_rbf, h0);

    // Stage 2: fused node MLP (WMMA fp32)
    const int    nblocks = (N_NODE_C + MLP_ROWS - 1) / MLP_ROWS;
    const size_t smem    = (size_t)2 * MLP_ROWS * LDH * sizeof(float);  // ~130 KB
    node_mlp<<<nblocks, 256, smem, stream>>>(h0, W_up, W1, b1, W2, b2, W3, b3, W_out, out);
}